// BloomFilter_6493990552263
// MI455X (gfx1250) — compile-verified
//
#include <hip/hip_runtime.h>

// Bloom filter for MI455X (gfx1250, wave32).
//
// Hash structure: positions_k = (v*PRIME + k) & (2^27-1), k=0..6 are 7
// CONSECUTIVE bits mod 2^27 starting at base = ((u32)v * PRIME) & MASK.
// So add = 1 (sometimes 2) dword atomic-ORs; query = 1 (sometimes 2) dword
// gathers + a 7-bit window test. All hash math is 32-bit.
//
// Cache policy: 16 MiB bitset lives in d_ws and stays L2-resident (192 MB L2).
// Streamed inputs/outputs (~128 MB single-use) get non-temporal hints so they
// don't evict the bitset.

typedef unsigned int       u32;
typedef unsigned long long u64;
typedef u32   v4u __attribute__((ext_vector_type(4)));
typedef float v4f __attribute__((ext_vector_type(4)));

static constexpr u32 kNumBits  = 1u << 27;          // 2^27 bits = 16 MiB
static constexpr u32 kBitMask  = kNumBits - 1u;
static constexpr u32 kNumWords = kNumBits >> 5;     // 4,194,304 dwords
static constexpr u32 kWordMask = kNumWords - 1u;
static constexpr u32 kPrime    = 2654435761u;
static constexpr u32 kSpan     = 0x7Fu;             // 7 consecutive hash bits

__device__ __forceinline__ u32 hash_base(u32 v) {
    return (v * kPrime) & kBitMask;
}

// ---------------------------------------------------------------- clear ----
// Regular-temporal stores: we WANT the zeroed bitset resident in L2 for the
// atomic phase that follows.
__global__ void __launch_bounds__(256) bloom_clear(v4u* __restrict__ bits) {
    u32 i = blockIdx.x * 256u + threadIdx.x;
    v4u z = {0u, 0u, 0u, 0u};
    bits[i] = z;
}

// ------------------------------------------------------------------ add ----
__device__ __forceinline__ void add_one(u32 v, u32* __restrict__ bits) {
    u32 base = hash_base(v);
    u32 w    = base >> 5;
    u32 b    = base & 31u;
    u64 m    = (u64)kSpan << b;                 // 7-bit span, may cross a dword
    atomicOr(&bits[w], (u32)m);                 // global_atomic_or_b32 (no ret)
    u32 hi = (u32)(m >> 32);
    if (hi)                                     // ~19% of lanes (EXEC-masked)
        atomicOr(&bits[(w + 1u) & kWordMask], hi);
}

__global__ void __launch_bounds__(256) bloom_add(const u32* __restrict__ vals,
                                                 u32* __restrict__ bits, int n) {
    int i = (int)(blockIdx.x * 256u + threadIdx.x) * 4;
    if (i + 3 < n) {
        // Two NT b128 loads cover 4 int64 values; low dwords are .x/.z.
        const v4u* p = (const v4u*)(vals + 2u * (u32)i);
        v4u a = __builtin_nontemporal_load(p);
        v4u c = __builtin_nontemporal_load(p + 1);
        add_one(a.x, bits);
        add_one(a.z, bits);
        add_one(c.x, bits);
        add_one(c.z, bits);
    } else {
        for (; i < n; ++i) add_one(vals[2u * (u32)i], bits);
    }
}

// ---------------------------------------------------------------- query ----
__device__ __forceinline__ float query_one(u32 v, const u32* __restrict__ bits) {
    u32 base = hash_base(v);
    u32 w    = base >> 5;
    u32 b    = base & 31u;
    u32 lo   = bits[w];                          // RT load: bitset stays in L2
    u32 hi   = 0u;
    if (b > 25u)                                 // span crosses dword boundary
        hi = bits[(w + 1u) & kWordMask];
    u64 cat = ((u64)hi << 32) | (u64)lo;
    return (((u32)(cat >> b) & kSpan) == kSpan) ? 1.0f : 0.0f;
}

__global__ void __launch_bounds__(256) bloom_query(const u32* __restrict__ vals,
                                                   const u32* __restrict__ bits,
                                                   float* __restrict__ out, int n) {
    int i = (int)(blockIdx.x * 256u + threadIdx.x) * 4;
    if (i + 3 < n) {
        const v4u* p = (const v4u*)(vals + 2u * (u32)i);
        v4u a = __builtin_nontemporal_load(p);
        v4u c = __builtin_nontemporal_load(p + 1);
        v4f r;
        r.x = query_one(a.x, bits);
        r.y = query_one(a.z, bits);
        r.z = query_one(c.x, bits);
        r.w = query_one(c.z, bits);
        __builtin_nontemporal_store(r, (v4f*)(out + i));   // NT b128 store
    } else {
        for (; i < n; ++i) out[i] = query_one(vals[2u * (u32)i], bits);
    }
}

// ----------------------------------------------------------------- host ----
extern "C" void kernel_launch(void* const* d_in, const int* in_sizes, int n_in,
                              void* d_out, int out_size, void* d_ws, size_t ws_size,
                              hipStream_t stream) {
    const u32* add_vals   = (const u32*)d_in[0];   // int64 viewed as dword pairs
    const u32* query_vals = (const u32*)d_in[1];
    int n_add = in_sizes[0];                       // 4,000,000
    int n_q   = in_sizes[1];                       // 8,000,000
    u32* bits = (u32*)d_ws;                        // 16 MiB bitset in scratch
    (void)ws_size; (void)n_in; (void)out_size;

    // 1) zero the bitset (deterministic every call; d_ws is poisoned once)
    bloom_clear<<<dim3(kNumWords / 4u / 256u), dim3(256), 0, stream>>>((v4u*)bits);

    // 2) scatter: 1-2 atomic ORs per value
    u32 add_threads = (u32)((n_add + 3) / 4);
    bloom_add<<<dim3((add_threads + 255u) / 256u), dim3(256), 0, stream>>>(
        add_vals, bits, n_add);

    // 3) gather + AND-reduce -> float 0/1
    u32 q_threads = (u32)((n_q + 3) / 4);
    bloom_query<<<dim3((q_threads + 255u) / 256u), dim3(256), 0, stream>>>(
        query_vals, bits, (float*)d_out, n_q);
}